// OptimizedLSTM_9998683865861
// MI455X (gfx1250) — compile-verified
//
#include <hip/hip_runtime.h>
#include <math.h>

// ---------------------------------------------------------------------------
// Fully-fused LSTM text classifier for MI455X (gfx1250, wave32).
//
//   out = concat(h[len-1], mean_t(emb), max_t(emb)) @ W_lin^T + b_lin
//
// One kernel, grid = B/16 workgroups of 128 threads (4 wave32 waves).
// Each workgroup owns 16 batch rows for all 512 timesteps:
//   - weights [W_ih^T ; W_hh^T] (100 x 256 zero-padded fp32) live in LDS,
//     stored PRE-PAIRED as float2 {Wcat[k][n], Wcat[k+1][n]} so one aligned
//     ds_load_b64 produces a ready WMMA B-fragment (no repack movs),
//   - per step: gather 16x50 embeddings (table is L2-resident: 80MB < 192MB L2),
//     g = [emb|h] @ Wcat via V_WMMA_F32_16X16X4_F32, 25 K-steps x 16 N-tiles,
//     exactly 4 tiles per wave (wave-uniform: EXEC all-1s, no saveexec),
//     gate nonlinearities + c/h update in LDS, capture h at t==len-1,
//     accumulate sum/max pooling of emb in LDS.
//   - epilogue: 16x5 output dots from LDS features.
// No intermediate tensor (xp: 105MB, h_all: 26MB) ever touches HBM.
// ---------------------------------------------------------------------------

typedef __attribute__((ext_vector_type(2))) float v2f;
typedef __attribute__((ext_vector_type(8))) float v8f;

#define Bc 256
#define Tc 512
#define Dc 50      // embedding dim
#define Hc 50      // lstm hidden
#define G4 200     // 4*H gate width
#define Cc 5       // classes
#define KC 100     // concatenated K = D + H
#define WSLDS 256  // padded N (16 tiles of 16)
#define NP (KC/4*2*WSLDS)  // # float2 weight pairs = 25*2*256 = 12800
#define ASLDS 102  // A-buffer row stride (even: b64-aligned; 102%64 banks OK)
#define GSLDS 257  // gate-buffer row stride (odd, >= 256)
#define SPOOL 51   // pooled/state row stride

#define SMEM_FLOATS (2*NP + 16*ASLDS + 16*GSLDS + 4*16*SPOOL + G4 + 16)

__global__ __launch_bounds__(128) void lstm_fused_kernel(
    const int*   __restrict__ x,       // [B,T]
    const int*   __restrict__ lengths, // [B]
    const float* __restrict__ emb,     // [V,D]
    const float* __restrict__ W_ih,    // [4H,D]
    const float* __restrict__ W_hh,    // [4H,H]
    const float* __restrict__ b_ih,    // [4H]
    const float* __restrict__ b_hh,    // [4H]
    const float* __restrict__ W_lin,   // [C,3H]
    const float* __restrict__ b_lin,   // [C]
    float*       __restrict__ out)     // [B,C]
{
  extern __shared__ float smem[];
  v2f*   Wp    = (v2f*)smem;              // [25][2][WSLDS] paired weights
  float* Abuf  = smem  + 2*NP;            // [16][ASLDS]  A = [emb_t | h]
  float* gbuf  = Abuf  + 16*ASLDS;        // [16][GSLDS]  pre-activation gates
  float* ssum  = gbuf  + 16*GSLDS;        // [16][SPOOL]  running sum(emb)
  float* smax  = ssum  + 16*SPOOL;        // [16][SPOOL]  running max(emb)
  float* scst  = smax  + 16*SPOOL;        // [16][SPOOL]  cell state c
  float* slast = scst  + 16*SPOOL;        // [16][SPOOL]  h at t = len-1
  float* biasg = slast + 16*SPOOL;        // [200]        b_ih + b_hh
  int*   lm1   = (int*)(biasg + G4);      // [16]         len-1 per row

  const int tid  = threadIdx.x;
  const int lane = tid & 31;
  const int wave = tid >> 5;
  const int b0   = blockIdx.x * 16;

  // ---- prologue: paired weights, bias, state init -----------------------
  // Wp[(kk*2+half)*256 + n] = { Wcat[4kk+2*half][n], Wcat[4kk+2*half+1][n] }
  for (int pidx = tid; pidx < NP; pidx += 128) {
    int kk   = pidx >> 9;            // / (2*WSLDS)
    int rem  = pidx & 511;
    int half = rem >> 8;
    int n    = rem & 255;
    int k0   = kk*4 + half*2;
    v2f v; v.x = 0.f; v.y = 0.f;
    if (n < G4) {
      v.x = (k0     < Dc) ? W_ih[n*Dc + k0]     : W_hh[n*Hc + (k0 - Dc)];
      v.y = (k0 + 1 < Dc) ? W_ih[n*Dc + k0 + 1] : W_hh[n*Hc + (k0 + 1 - Dc)];
    }
    Wp[pidx] = v;
  }
  for (int idx = tid; idx < G4; idx += 128) biasg[idx] = b_ih[idx] + b_hh[idx];
  for (int e = tid; e < 16*Hc; e += 128) {
    int b = e / Hc, j = e - b*Hc;
    ssum[b*SPOOL + j]  = 0.f;
    smax[b*SPOOL + j]  = -INFINITY;
    scst[b*SPOOL + j]  = 0.f;
    slast[b*SPOOL + j] = 0.f;
    Abuf[b*ASLDS + Dc + j] = 0.f;        // h0 = 0
  }
  if (tid < 16) {
    int L = lengths[b0 + tid];
    lm1[tid] = (L < Tc ? L : Tc) - 1;
  }

  const int m     = lane & 15;           // matrix row / column-in-tile
  const int half  = lane >> 4;           // lanes 0-15: K+0/1, lanes 16-31: K+2/3
  const int mrow  = half << 3;           // C/D: lanes 0-15 -> M 0-7, 16-31 -> M 8-15
  const int nbase = wave*64 + m;         // wave owns columns [wave*64, wave*64+64)
  const v2f* __restrict__ Arow = (const v2f*)(Abuf + m*ASLDS) + half; // + koff/2

  // ---- sequential scan over T -------------------------------------------
  for (int t = 0; t < Tc; ++t) {
    __syncthreads();                     // prev gate-phase done before gather

    // gather embeddings for this timestep + pooling accumulation
    for (int e = tid; e < 16*Dc; e += 128) {
      int b = e / Dc, j = e - b*Dc;
      int tok = x[(size_t)(b0 + b)*Tc + t];
      float v = emb[(size_t)tok*Dc + j];
      Abuf[b*ASLDS + j] = v;
      ssum[b*SPOOL + j] += v;
      smax[b*SPOOL + j]  = fmaxf(smax[b*SPOOL + j], v);
    }
    // warm L2/L0 for the next step's embedding rows (global_prefetch_b8)
    if (t + 1 < Tc && tid < 16) {
      int tok2 = x[(size_t)(b0 + tid)*Tc + t + 1];
      __builtin_prefetch(&emb[(size_t)tok2*Dc], 0, 0);
      __builtin_prefetch(&emb[(size_t)tok2*Dc + 32], 0, 0);
    }
    __syncthreads();

    // WMMA: g[16 x 256] = A[16 x 100] @ Wcat[100 x 256]
    // wave w owns N-tiles {4w .. 4w+3}: uniform count -> EXEC all-1s
    v8f acc0, acc1, acc2, acc3;
    #pragma unroll
    for (int r = 0; r < 8; ++r) { acc0[r]=0.f; acc1[r]=0.f; acc2[r]=0.f; acc3[r]=0.f; }

    for (int kk = 0; kk < 25; ++kk) {
      v2f a = Arow[kk*2];                               // {A[m][koff], A[m][koff+1]}
      const v2f* wr = Wp + (kk*2 + half)*WSLDS + nbase; // pre-paired B-fragments
      acc0 = __builtin_amdgcn_wmma_f32_16x16x4_f32(false, a, false, wr[ 0], (short)0, acc0, false, false);
      acc1 = __builtin_amdgcn_wmma_f32_16x16x4_f32(false, a, false, wr[16], (short)0, acc1, false, false);
      acc2 = __builtin_amdgcn_wmma_f32_16x16x4_f32(false, a, false, wr[32], (short)0, acc2, false, false);
      acc3 = __builtin_amdgcn_wmma_f32_16x16x4_f32(false, a, false, wr[48], (short)0, acc3, false, false);
    }

    // spill gate tiles to LDS (padded columns >=200 are inert zeros)
    #pragma unroll
    for (int r = 0; r < 8; ++r) {
      float* grow = gbuf + (mrow + r)*GSLDS + nbase;
      grow[ 0] = acc0[r];
      grow[16] = acc1[r];
      grow[32] = acc2[r];
      grow[48] = acc3[r];
    }
    __syncthreads();

    // gate nonlinearities + state update (PyTorch gate order i,f,g,o)
    for (int e = tid; e < 16*Hc; e += 128) {
      int b = e / Hc, j = e - b*Hc;
      float gi = gbuf[b*GSLDS + j         ] + biasg[j         ];
      float gf = gbuf[b*GSLDS + j +   Hc  ] + biasg[j +   Hc  ];
      float gg = gbuf[b*GSLDS + j + 2*Hc  ] + biasg[j + 2*Hc  ];
      float go = gbuf[b*GSLDS + j + 3*Hc  ] + biasg[j + 3*Hc  ];
      float i_ = 1.f / (1.f + __expf(-gi));
      float f_ = 1.f / (1.f + __expf(-gf));
      float o_ = 1.f / (1.f + __expf(-go));
      float c_ = f_ * scst[b*SPOOL + j] + i_ * tanhf(gg);
      float h_ = o_ * tanhf(c_);
      scst[b*SPOOL + j] = c_;
      Abuf[b*ASLDS + Dc + j] = h_;       // feeds next step's A-matrix
      if (t == lm1[b]) slast[b*SPOOL + j] = h_;
    }
  }
  __syncthreads();

  // ---- epilogue: out[b][c] = [last, sum/len, max] . W_lin[c] + b_lin[c] --
  if (tid < 16*Cc) {
    int b = tid / Cc, c = tid - b*Cc;
    float invL = 1.f / (float)(lm1[b] + 1);
    float accv = b_lin[c];
    const float* wl = W_lin + c*(3*Hc);
    #pragma unroll 5
    for (int j = 0; j < Hc; ++j) {
      accv += slast[b*SPOOL + j]        * wl[j];
      accv += ssum[b*SPOOL + j] * invL  * wl[Hc + j];
      accv += smax[b*SPOOL + j]         * wl[2*Hc + j];
    }
    out[(b0 + b)*Cc + c] = accv;
  }
}

extern "C" void kernel_launch(void* const* d_in, const int* in_sizes, int n_in,
                              void* d_out, int out_size, void* d_ws, size_t ws_size,
                              hipStream_t stream) {
  const int*   x       = (const int*)  d_in[0];
  const int*   lengths = (const int*)  d_in[1];
  // d_in[2] = bows, unused by the model
  const float* emb     = (const float*)d_in[3];
  const float* W_ih    = (const float*)d_in[4];
  const float* W_hh    = (const float*)d_in[5];
  const float* b_ih    = (const float*)d_in[6];
  const float* b_hh    = (const float*)d_in[7];
  const float* W_lin   = (const float*)d_in[8];
  const float* b_lin   = (const float*)d_in[9];
  float* out = (float*)d_out;

  const size_t smemBytes = (size_t)SMEM_FLOATS * sizeof(float); // ~139 KB < 160 KB
  lstm_fused_kernel<<<Bc/16, 128, smemBytes, stream>>>(
      x, lengths, emb, W_ih, W_hh, b_ih, b_hh, W_lin, b_lin, out);
}